// DIAGCN_34677565948517
// MI455X (gfx1250) — compile-verified
//
#include <hip/hip_runtime.h>
#include <hip/hip_bf16.h>

typedef __attribute__((ext_vector_type(2))) float v2f;
typedef __attribute__((ext_vector_type(8))) float v8f;

#define IN_DIM   1024
#define HID      512
#define ZSTR     1536   // h | z0(agg) | z1
#define TM       128
#define TN       64
#define TK       32
#define XS_STRIDE 36    // 32 + 4 pad: float4-aligned, conflict-free 16-lane row reads
#define WS_STRIDE 68    // 64 + 4 pad

// ---------------- utility kernels ----------------

__global__ void zero_f32(float* __restrict__ p, int n) {
    int i = blockIdx.x * blockDim.x + threadIdx.x;
    if (i < n) p[i] = 0.0f;
}

__global__ void count_edges(const int* __restrict__ edges, const int* __restrict__ rel,
                            float* __restrict__ cnt, int E, int N) {
    int e = blockIdx.x * blockDim.x + threadIdx.x;
    if (e < E) {
        int t = edges[E + e];
        int r = rel[e];
        atomicAdd(&cnt[(size_t)r * N + t], 1.0f);
    }
}

__global__ void invert_counts(float* __restrict__ c, int n) {
    int i = blockIdx.x * blockDim.x + threadIdx.x;
    if (i < n) c[i] = 1.0f / fmaxf(c[i], 1.0f);
}

// Fold the post-h linear algebra into 7-wide matrices:
// A = w_gc_rel@w_clf, B = w_gc_root@w_clf, C = w_skip@w_clf, d = (b_gc+b_skip)@w_clf + b_clf
__global__ void fuse_weights(const float* __restrict__ w_gc_rel, const float* __restrict__ w_gc_root,
                             const float* __restrict__ w_skip, const float* __restrict__ w_clf,
                             const float* __restrict__ b_gc, const float* __restrict__ b_skip,
                             const float* __restrict__ b_clf,
                             float* __restrict__ Af, float* __restrict__ Bf,
                             float* __restrict__ Cf, float* __restrict__ dv) {
    int idx = blockIdx.x * blockDim.x + threadIdx.x;
    if (idx < 3584) {                       // A: [512,7]
        int i = idx / 7, j = idx % 7; float s = 0.f;
        for (int k = 0; k < HID; ++k) s += w_gc_rel[i * HID + k] * w_clf[k * 7 + j];
        Af[idx] = s;
    } else if (idx < 7168) {                // B: [512,7]
        int t = idx - 3584; int i = t / 7, j = t % 7; float s = 0.f;
        for (int k = 0; k < HID; ++k) s += w_gc_root[i * HID + k] * w_clf[k * 7 + j];
        Bf[t] = s;
    } else if (idx < 14336) {               // C: [1024,7]
        int t = idx - 7168; int i = t / 7, j = t % 7; float s = 0.f;
        for (int k = 0; k < HID; ++k) s += w_skip[i * HID + k] * w_clf[k * 7 + j];
        Cf[t] = s;
    } else if (idx < 14343) {               // d: [7]
        int j = idx - 14336; float s = b_clf[j];
        for (int k = 0; k < HID; ++k) s += (b_gc[k] + b_skip[k]) * w_clf[k * 7 + j];
        dv[j] = s;
    }
}

// ---------------- main fp32 WMMA GEMM: Z[N,1536] = X[N,1024] @ [w_root|w_rel0|w_rel1] ----------------
// h columns (0..511) get +b_rgcn.

__global__ __launch_bounds__(256) void gemm_wmma(const float* __restrict__ x,
                                                 const float* __restrict__ w_root,
                                                 const float* __restrict__ w_rel,
                                                 const float* __restrict__ b_rgcn,
                                                 float* __restrict__ Z, int N) {
    __shared__ __align__(16) float Xs[TM * XS_STRIDE];
    __shared__ __align__(16) float Ws[TK * WS_STRIDE];

    const int tid  = threadIdx.x;
    const int lane = tid & 31;
    const int wv   = tid >> 5;        // 0..7 waves
    const int half = lane >> 4;
    const int l16  = lane & 15;

    const int m0 = blockIdx.x * TM;
    const int c0 = blockIdx.y * TN;   // global column base, within one weight matrix (64 | 512)

    const float* wsrc; int cofs; bool addBias;
    if (c0 < 512)       { wsrc = w_root;                 cofs = c0;        addBias = true;  }
    else if (c0 < 1024) { wsrc = w_rel;                  cofs = c0 - 512;  addBias = false; }
    else                { wsrc = w_rel + 1024 * 512;     cofs = c0 - 1024; addBias = false; }

    const int wr0 = (wv & 3) * 32;    // wave row offset within block tile
    const int wc0 = (wv >> 2) * 32;   // wave col offset

    v8f acc[2][2] = {};

    // global->LDS mappings
    const int xlr = tid >> 3;            // 0..31 rows per pass (x4 passes)
    const int xlc = (tid & 7) * 4;       // 0..28 step 4
    const int wlr = tid >> 4;            // 0..15 rows per pass (x2 passes)
    const int wlc = (tid & 15) * 4;      // 0..60 step 4

    for (int kt = 0; kt < IN_DIM; kt += TK) {
        // stage X tile [128 x 32]
        #pragma unroll
        for (int r = 0; r < 4; ++r) {
            int row = xlr + 32 * r;
            int gr  = m0 + row;
            float4 v = make_float4(0.f, 0.f, 0.f, 0.f);
            if (gr < N) v = *(const float4*)(x + (size_t)gr * IN_DIM + kt + xlc);
            *(float4*)&Xs[row * XS_STRIDE + xlc] = v;
        }
        // stage W tile [32 x 64]
        #pragma unroll
        for (int r = 0; r < 2; ++r) {
            int row = wlr + 16 * r;
            float4 v = *(const float4*)(wsrc + (size_t)(kt + row) * HID + cofs + wlc);
            *(float4*)&Ws[row * WS_STRIDE + wlc] = v;
        }
        __syncthreads();

        if (kt + TK < IN_DIM) { // prefetch next tiles (global_prefetch_b8)
            int gr = m0 + xlr;
            if (gr < N) __builtin_prefetch(x + (size_t)gr * IN_DIM + kt + TK + xlc, 0, 1);
            __builtin_prefetch(wsrc + (size_t)(kt + TK + wlr) * HID + cofs + wlc, 0, 1);
        }

        #pragma unroll
        for (int k = 0; k < TK; k += 4) {
            const int ka = k + 2 * half;  // lanes 0-15: K k,k+1 ; lanes 16-31: K k+2,k+3
            v2f a0, a1, b0, b1;
            a0.x = Xs[(wr0 + l16) * XS_STRIDE + ka];
            a0.y = Xs[(wr0 + l16) * XS_STRIDE + ka + 1];
            a1.x = Xs[(wr0 + 16 + l16) * XS_STRIDE + ka];
            a1.y = Xs[(wr0 + 16 + l16) * XS_STRIDE + ka + 1];
            b0.x = Ws[ka * WS_STRIDE + wc0 + l16];
            b0.y = Ws[(ka + 1) * WS_STRIDE + wc0 + l16];
            b1.x = Ws[ka * WS_STRIDE + wc0 + 16 + l16];
            b1.y = Ws[(ka + 1) * WS_STRIDE + wc0 + 16 + l16];
            acc[0][0] = __builtin_amdgcn_wmma_f32_16x16x4_f32(false, a0, false, b0, (short)0, acc[0][0], false, false);
            acc[0][1] = __builtin_amdgcn_wmma_f32_16x16x4_f32(false, a0, false, b1, (short)0, acc[0][1], false, false);
            acc[1][0] = __builtin_amdgcn_wmma_f32_16x16x4_f32(false, a1, false, b0, (short)0, acc[1][0], false, false);
            acc[1][1] = __builtin_amdgcn_wmma_f32_16x16x4_f32(false, a1, false, b1, (short)0, acc[1][1], false, false);
        }
        __syncthreads();
    }

    // epilogue: C/D layout — VGPR v, lanes<16 -> M=v, lanes>=16 -> M=v+8; N = l16
    #pragma unroll
    for (int i = 0; i < 2; ++i) {
        #pragma unroll
        for (int j = 0; j < 2; ++j) {
            int gcb  = c0 + wc0 + 16 * j + l16;
            float bias = addBias ? b_rgcn[gcb] : 0.0f;
            #pragma unroll
            for (int v = 0; v < 8; ++v) {
                int gr = m0 + wr0 + 16 * i + v + 8 * half;
                if (gr < N) Z[(size_t)gr * ZSTR + gcb] = acc[i][j][v] + bias;
            }
        }
    }
}

// ---------------- graph scatters ----------------

// h[tgt] += z_rel[src] * (1/max(count,1))  — one block per edge, 2 cols/thread
__global__ void scatter_mean(const int* __restrict__ edges, const int* __restrict__ rel,
                             const float* __restrict__ cinv, float* __restrict__ Z,
                             int E, int N) {
    int e = blockIdx.x;
    int s = edges[e];
    int t = edges[E + e];
    int r = rel[e];
    float sc = cinv[(size_t)r * N + t];
    const float* zsrc = Z + (size_t)s * ZSTR + 512 + r * 512;
    float* hdst = Z + (size_t)t * ZSTR;
    int c = threadIdx.x * 2;
    float2 v = *(const float2*)(zsrc + c);
    atomicAdd(hdst + c,     v.x * sc);
    atomicAdd(hdst + c + 1, v.y * sc);
}

// zero the agg region (reuses consumed z0 columns 512..1023), float4 per thread
__global__ void zero_agg(float* __restrict__ Z, int N) {
    int i = blockIdx.x * blockDim.x + threadIdx.x;
    if (i < N * 128) {
        int row = i >> 7;
        int col = (i & 127) * 4;
        *(float4*)&Z[(size_t)row * ZSTR + 512 + col] = make_float4(0.f, 0.f, 0.f, 0.f);
    }
}

// agg[tgt] += h[src]
__global__ void agg_scatter(const int* __restrict__ edges, float* __restrict__ Z, int E) {
    int e = blockIdx.x;
    int s = edges[e];
    int t = edges[E + e];
    const float* hsrc = Z + (size_t)s * ZSTR;
    float* adst = Z + (size_t)t * ZSTR + 512;
    int c = threadIdx.x * 2;
    float2 v = *(const float2*)(hsrc + c);
    atomicAdd(adst + c,     v.x);
    atomicAdd(adst + c + 1, v.y);
}

// ---------------- fused classifier: out = agg@A + h@B + x@C + d ----------------
// one wave per node; lanes stride K, wave32 butterfly reduce
__global__ __launch_bounds__(256) void final_kernel(const float* __restrict__ x,
                                                    const float* __restrict__ Z,
                                                    const float* __restrict__ Af, const float* __restrict__ Bf,
                                                    const float* __restrict__ Cf, const float* __restrict__ dv,
                                                    float* __restrict__ out, int N) {
    int wv   = threadIdx.x >> 5;
    int lane = threadIdx.x & 31;
    int node = blockIdx.x * 8 + wv;
    if (node >= N) return;
    const float* h   = Z + (size_t)node * ZSTR;
    const float* agg = h + 512;
    float acc[7] = {0.f, 0.f, 0.f, 0.f, 0.f, 0.f, 0.f};
    for (int k = lane; k < HID; k += 32) {
        float hv = h[k], av = agg[k];
        const float* ar = Af + k * 7;
        const float* br = Bf + k * 7;
        #pragma unroll
        for (int j = 0; j < 7; ++j) acc[j] += av * ar[j] + hv * br[j];
    }
    const float* xr = x + (size_t)node * IN_DIM;
    for (int k = lane; k < IN_DIM; k += 32) {
        float xv = xr[k];
        const float* cr = Cf + k * 7;
        #pragma unroll
        for (int j = 0; j < 7; ++j) acc[j] += xv * cr[j];
    }
    #pragma unroll
    for (int j = 0; j < 7; ++j) {
        #pragma unroll
        for (int off = 16; off > 0; off >>= 1) acc[j] += __shfl_xor(acc[j], off, 32);
    }
    if (lane == 0) {
        #pragma unroll
        for (int j = 0; j < 7; ++j) out[(size_t)node * 7 + j] = acc[j] + dv[j];
    }
}

// ---------------- host launcher ----------------

extern "C" void kernel_launch(void* const* d_in, const int* in_sizes, int n_in,
                              void* d_out, int out_size, void* d_ws, size_t ws_size,
                              hipStream_t stream) {
    const float* x        = (const float*)d_in[0];
    const int*   edges    = (const int*)d_in[1];
    const int*   rel      = (const int*)d_in[2];
    const float* w_rel    = (const float*)d_in[3];
    const float* w_root   = (const float*)d_in[4];
    const float* b_rgcn   = (const float*)d_in[5];
    const float* w_gc_rel = (const float*)d_in[6];
    const float* w_gc_root= (const float*)d_in[7];
    const float* b_gc     = (const float*)d_in[8];
    const float* w_skip   = (const float*)d_in[9];
    const float* b_skip   = (const float*)d_in[10];
    const float* w_clf    = (const float*)d_in[11];
    const float* b_clf    = (const float*)d_in[12];
    float* out = (float*)d_out;

    const int N = in_sizes[0] / IN_DIM;   // 50000
    const int E = in_sizes[2];            // 245000

    // workspace layout (floats): Z[N*1536] | cinv[2N] | A[3584] | B[3584] | C[7168] | d[7]
    float* Z    = (float*)d_ws;
    float* cinv = Z + (size_t)N * ZSTR;
    float* Af   = cinv + 2 * (size_t)N;
    float* Bf   = Af + 3584;
    float* Cf   = Bf + 3584;
    float* dv   = Cf + 7168;

    zero_f32<<<(2 * N + 255) / 256, 256, 0, stream>>>(cinv, 2 * N);
    count_edges<<<(E + 255) / 256, 256, 0, stream>>>(edges, rel, cinv, E, N);
    invert_counts<<<(2 * N + 255) / 256, 256, 0, stream>>>(cinv, 2 * N);
    fuse_weights<<<(14343 + 255) / 256, 256, 0, stream>>>(w_gc_rel, w_gc_root, w_skip, w_clf,
                                                          b_gc, b_skip, b_clf, Af, Bf, Cf, dv);
    dim3 g((N + TM - 1) / TM, ZSTR / TN);
    gemm_wmma<<<g, 256, 0, stream>>>(x, w_root, w_rel, b_rgcn, Z, N);
    scatter_mean<<<E, 256, 0, stream>>>(edges, rel, cinv, Z, E, N);
    zero_agg<<<(N * 128 + 255) / 256, 256, 0, stream>>>(Z, N);
    agg_scatter<<<E, 256, 0, stream>>>(edges, Z, E);
    final_kernel<<<(N + 7) / 8, 256, 0, stream>>>(x, Z, Af, Bf, Cf, dv, out, N);
}